// MultiBoxLoss_43198781063465
// MI455X (gfx1250) — compile-verified
//
#include <hip/hip_runtime.h>
#include <hip/hip_bf16.h>
#include <math.h>

#ifndef __has_builtin
#define __has_builtin(x) 0
#endif

#define B_   64
#define P_   8732
#define G_   16
#define C_   81
#define NBP  558848   // B_*P_
#define THRESH_ 0.5f

typedef float v2f __attribute__((ext_vector_type(2)));
typedef float v8f __attribute__((ext_vector_type(8)));
typedef int   v4i __attribute__((ext_vector_type(4)));
typedef __attribute__((address_space(1))) v4i gv4i;
typedef __attribute__((address_space(3))) v4i lv4i;

// ---------------------------------------------------------------------------
// Exact f32 wave32 sum on the matrix pipe (v_wmma_f32_16x16x4_f32):
//   A[16x4] holds the 32 lane values (K=0 for lanes 0-15, K=2 for 16-31),
//   B[4x16] = ones  =>  D[m,n] = v_m + v_{m+16}  for every column n.
//   Per-lane sum of the 8 D regs + shfl_xor(16) = full 32-lane sum.
// Requires EXEC == all ones at the call site.
// ---------------------------------------------------------------------------
__device__ __forceinline__ float wave_sum_wmma(float v) {
  v2f a;  a[0] = v;    a[1] = 0.0f;
  v2f bo; bo[0] = 1.0f; bo[1] = 1.0f;
  v8f c = {0.f, 0.f, 0.f, 0.f, 0.f, 0.f, 0.f, 0.f};
  v8f d = __builtin_amdgcn_wmma_f32_16x16x4_f32(false, a, false, bo,
                                                (short)0, c, false, false);
  float s = d[0] + d[1] + d[2] + d[3] + d[4] + d[5] + d[6] + d[7];
  s += __shfl_xor(s, 16, 32);
  return s;
}

__device__ __forceinline__ float wave_max32(float v) {
  #pragma unroll
  for (int o = 16; o > 0; o >>= 1) v = fmaxf(v, __shfl_xor(v, o, 32));
  return v;
}

// ---------------------------------------------------------------------------
// gfx1250 async global->LDS copy (ASYNCcnt-tracked), 16B per lane.
// Builtin signature (per hipcc diagnostic): (int4 AS1*, int4 AS3*, Ii, Ii).
// Falls back to a synchronous float4 copy if the builtin is unavailable.
// ---------------------------------------------------------------------------
__device__ __forceinline__ void async_copy16(const float* src, float* ldsDst) {
#if __has_builtin(__builtin_amdgcn_global_load_async_to_lds_b128)
  __builtin_amdgcn_global_load_async_to_lds_b128(
      (gv4i*)(v4i*)src, (lv4i*)(v4i*)ldsDst, 0, 0);
#else
  *(float4*)ldsDst = *(const float4*)src;
#endif
}

__device__ __forceinline__ void async_wait0() {
#if __has_builtin(__builtin_amdgcn_s_wait_asynccnt)
  __builtin_amdgcn_s_wait_asynccnt(0);
#endif
}

// ---------------------------------------------------------------------------
// Zero accumulators (harness does not re-poison between graph replays).
// ---------------------------------------------------------------------------
__global__ void k_init(int* num_pos, float* accum) {
  int t = threadIdx.x;
  if (t < B_) num_pos[t] = 0;
  if (t < 3)  accum[t]  = 0.0f;
}

// ---------------------------------------------------------------------------
// Per-batch matching: best IoU / best gt per prior, best prior per gt.
// One block per batch; gt boxes cached in LDS; per-gt argmax via packed u64
// atomicMax (iou bits high, ~prior_idx low => smallest-index tie-break).
// ---------------------------------------------------------------------------
__global__ void k_assign(const float* __restrict__ priors,
                         const float* __restrict__ gt_box,
                         float* __restrict__ best_iou,
                         int*   __restrict__ best_g,
                         int*   __restrict__ best_prior) {
  const int b   = blockIdx.x;
  const int tid = threadIdx.x;
  __shared__ float4 sg[G_];
  __shared__ unsigned long long sbest[G_];
  if (tid < G_) {
    sg[tid]    = ((const float4*)gt_box)[b * G_ + tid];
    sbest[tid] = 0ull;
  }
  __syncthreads();

  float    lbv[G_];
  unsigned lbp[G_];
  #pragma unroll
  for (int g = 0; g < G_; ++g) { lbv[g] = -1.0f; lbp[g] = 0u; }

  for (int p = tid; p < P_; p += blockDim.x) {
    float4 pr = ((const float4*)priors)[p];
    const float px1 = pr.x - pr.z * 0.5f, py1 = pr.y - pr.w * 0.5f;
    const float px2 = pr.x + pr.z * 0.5f, py2 = pr.y + pr.w * 0.5f;
    const float area2 = pr.z * pr.w;
    float best = -1.0f; int bg = 0;
    #pragma unroll
    for (int g = 0; g < G_; ++g) {
      float4 gb = sg[g];
      float ix = fmaxf(fminf(gb.z, px2) - fmaxf(gb.x, px1), 0.0f);
      float iy = fmaxf(fminf(gb.w, py2) - fmaxf(gb.y, py1), 0.0f);
      float inter = ix * iy;
      float a1 = (gb.z - gb.x) * (gb.w - gb.y);
      float iou = inter / (a1 + area2 - inter);
      if (iou > best) { best = iou; bg = g; }
      if (iou > lbv[g]) { lbv[g] = iou; lbp[g] = (unsigned)p; }
    }
    best_iou[b * P_ + p] = best;
    best_g[b * P_ + p]   = bg;
  }

  #pragma unroll
  for (int g = 0; g < G_; ++g) {
    unsigned long long pack =
        ((unsigned long long)__float_as_uint(fmaxf(lbv[g], 0.0f)) << 32) |
        (unsigned long long)(0xFFFFFFFFu - lbp[g]);
    atomicMax(&sbest[g], pack);
  }
  __syncthreads();
  if (tid < G_)
    best_prior[b * G_ + tid] =
        (int)(0xFFFFFFFFu - (unsigned)(sbest[tid] & 0xFFFFFFFFull));
}

// ---------------------------------------------------------------------------
// Force-keep, class targets, encode + masked smooth-L1 (WMMA wave reduction).
// Grid covers exactly NBP threads (NBP % 256 == 0 -> every wave full).
// ---------------------------------------------------------------------------
__global__ void k_encode(const float* __restrict__ loc_p,
                         const float* __restrict__ priors,
                         const float* __restrict__ gt_box,
                         const int*   __restrict__ gt_class,
                         const float* __restrict__ best_iou,
                         const int*   __restrict__ best_g,
                         const int*   __restrict__ best_prior,
                         int*   __restrict__ class_t,
                         int*   __restrict__ num_pos,
                         float* __restrict__ accum) {
  const int idx = blockIdx.x * blockDim.x + threadIdx.x;   // < NBP by construction
  const int b = idx / P_;
  const int p = idx - b * P_;

  float bi = best_iou[idx];
  const int bg = best_g[idx];
  #pragma unroll
  for (int g = 0; g < G_; ++g)
    if (best_prior[b * G_ + g] == p) bi = 2.0f;

  int cls = gt_class[b * G_ + bg] + 1;
  if (bi < THRESH_) cls = 0;
  class_t[idx] = cls;

  float sl1 = 0.0f;
  if (cls > 0) {
    atomicAdd(&num_pos[b], 1);
    float4 m  = ((const float4*)gt_box)[b * G_ + bg];
    float4 pr = ((const float4*)priors)[p];
    float4 lp = ((const float4*)loc_p)[idx];
    float t0 = ((m.x + m.z) * 0.5f - pr.x) / (pr.z * 0.1f);
    float t1 = ((m.y + m.w) * 0.5f - pr.y) / (pr.w * 0.1f);
    float t2 = logf((m.z - m.x) / pr.z) * 5.0f;   // / VARIANCE[1]=0.2
    float t3 = logf((m.w - m.y) / pr.w) * 5.0f;
    float dv[4] = {lp.x - t0, lp.y - t1, lp.z - t2, lp.w - t3};
    #pragma unroll
    for (int d = 0; d < 4; ++d) {
      float diff = fabsf(dv[d]);
      sl1 += (diff < 1.0f) ? 0.5f * diff * diff : diff - 0.5f;
    }
  }
  float wsum = wave_sum_wmma(sl1);
  if ((threadIdx.x & 31) == 0) atomicAdd(&accum[0], wsum);
}

// ---------------------------------------------------------------------------
// Cross entropy: one wave32 per (b,p) row of 81 classes.
// Lanes read c, c+32, c+64 (three coalesced strips of the 324B row).
// Softmax denominator reduced on the matrix pipe; target logit broadcast
// with a single shfl (t is wave-uniform) instead of a 5-stage max tree.
// 181 MB stream -> this kernel sets the runtime; keep it pure bandwidth.
// ---------------------------------------------------------------------------
__global__ void k_ce(const float* __restrict__ class_p,
                     const int*   __restrict__ class_t,
                     float* __restrict__ ce,
                     float* __restrict__ accum) {
  const int gthread = blockIdx.x * blockDim.x + threadIdx.x;
  const int w    = gthread >> 5;          // prior index, exact coverage
  const int lane = threadIdx.x & 31;
  const float* row = class_p + (size_t)w * C_;

  const int t = class_t[w];               // wave-uniform target class

  const float x0 = row[lane];
  const float x1 = row[lane + 32];
  const bool  has2 = (lane + 64) < C_;
  const float x2 = has2 ? row[lane + 64] : -3.4e38f;

  float m = wave_max32(fmaxf(fmaxf(x0, x1), x2));
  float s = expf(x0 - m) + expf(x1 - m) + (has2 ? expf(x2 - m) : 0.0f);
  float stot = wave_sum_wmma(s);

  // x[t]: slot and source lane are wave-uniform -> one broadcast shuffle
  const int   src = (t < 32) ? t : ((t < 64) ? t - 32 : t - 64);
  const float sel = (t < 32) ? x0 : ((t < 64) ? x1 : x2);
  const float xt  = __shfl(sel, src, 32);

  float cev = logf(stot) + m - xt;
  if (lane == 0) {
    ce[w] = cev;
    if (t > 0) atomicAdd(&accum[1], cev);   // positives are always selected
  }
}

// ---------------------------------------------------------------------------
// Hard-negative mining, one block per batch. Sum of top-K loss_gt is
// tie-invariant, so instead of argsort(argsort(.)) we bisect the K-th largest
// value on float bit patterns (loss_gt >= 0 -> uint order == float order):
//   ans = max u with count(x >= u) >= K;  top-K sum = sum_{x>ans} x + (K-c)*ans
// The 35KB per-batch CE slice is staged into LDS ONCE via the gfx1250 async
// global->LDS path, positives zeroed in place, then all 31 count passes and
// the final sum run out of LDS (no repeated L2 traffic).
// ---------------------------------------------------------------------------
__global__ void k_mine(const float* __restrict__ ce,
                       const int*   __restrict__ class_t,
                       const int*   __restrict__ num_pos,
                       float* __restrict__ accum) {
  const int b = blockIdx.x, tid = threadIdx.x;
  __shared__ __align__(16) float s_lgt[P_];   // 34928 B
  __shared__ int      sred[256];
  __shared__ float    sf[8];
  __shared__ unsigned s_ans;
  __shared__ int      sK;

  // stage ce[b,:] into LDS (2183 x 16B chunks; base is 16B aligned)
  const float* src = ce + (size_t)b * P_;
  for (int i = tid; i < P_ / 4; i += 256)
    async_copy16(src + 4 * i, &s_lgt[4 * i]);
  async_wait0();
  __syncthreads();

  // zero out positives: loss_gt = pos ? 0 : ce
  for (int p = tid; p < P_; p += 256)
    if (class_t[b * P_ + p] > 0) s_lgt[p] = 0.0f;

  if (tid == 0) {
    int K = 3 * num_pos[b];
    if (K > P_ - 1) K = P_ - 1;
    sK = K; s_ans = 0u;
  }
  __syncthreads();
  const int K = sK;
  unsigned ans = 0u;

  for (int bit = 30; bit >= 0; --bit) {
    const unsigned candu = ans | (1u << bit);
    int cnt = 0;
    for (int p = tid; p < P_; p += 256)
      cnt += (__float_as_uint(s_lgt[p]) >= candu) ? 1 : 0;
    sred[tid] = cnt; __syncthreads();
    #pragma unroll
    for (int s2 = 128; s2 > 0; s2 >>= 1) {
      if (tid < s2) sred[tid] += sred[tid + s2];
      __syncthreads();
    }
    if (tid == 0 && sred[0] >= K) s_ans = candu;
    __syncthreads();
    ans = s_ans;
  }

  float fsum = 0.0f; int cgt = 0;
  for (int p = tid; p < P_; p += 256) {
    const float v = s_lgt[p];
    if (__float_as_uint(v) > ans) { fsum += v; ++cgt; }
  }
  sred[tid] = cgt; __syncthreads();
  #pragma unroll
  for (int s2 = 128; s2 > 0; s2 >>= 1) {
    if (tid < s2) sred[tid] += sred[tid + s2];
    __syncthreads();
  }
  float wsum = wave_sum_wmma(fsum);          // full wave, matrix-pipe reduce
  if ((tid & 31) == 0) sf[tid >> 5] = wsum;
  __syncthreads();
  if (tid == 0) {
    float tot = 0.0f;
    #pragma unroll
    for (int i = 0; i < 8; ++i) tot += sf[i];
    if (K > 0) tot += (float)(K - sred[0]) * __uint_as_float(ans);
    else       tot  = 0.0f;
    atomicAdd(&accum[2], tot);
  }
}

// ---------------------------------------------------------------------------
__global__ void k_final(const int* __restrict__ num_pos,
                        const float* __restrict__ accum,
                        float* __restrict__ out) {
  if (blockIdx.x == 0 && threadIdx.x == 0) {
    int N = 0;
    #pragma unroll
    for (int b = 0; b < B_; ++b) N += num_pos[b];
    float n = (N > 0) ? (float)N : 1.0f;
    out[0] = accum[0] / n;
    out[1] = (accum[1] + accum[2]) / n;
  }
}

// ---------------------------------------------------------------------------
extern "C" void kernel_launch(void* const* d_in, const int* in_sizes, int n_in,
                              void* d_out, int out_size, void* d_ws, size_t ws_size,
                              hipStream_t stream) {
  (void)in_sizes; (void)n_in; (void)out_size; (void)ws_size;
  const float* loc_p    = (const float*)d_in[0];   // [B,P,4]
  const float* class_p  = (const float*)d_in[1];   // [B,P,C]
  const float* priors   = (const float*)d_in[2];   // [P,4]
  const float* gt_box   = (const float*)d_in[3];   // [B,G,4]
  const int*   gt_class = (const int*)  d_in[4];   // [B,G]
  float* out = (float*)d_out;

  // workspace (floats): best_iou | ce | class_t | best_g | best_prior | num_pos | accum[3]
  float* wsf        = (float*)d_ws;
  float* best_iou   = wsf;
  float* ce         = wsf + (size_t)NBP;
  int*   class_t    = (int*)(wsf + (size_t)2 * NBP);
  int*   best_g     = (int*)(wsf + (size_t)3 * NBP);
  int*   best_prior = (int*)(wsf + (size_t)4 * NBP);
  int*   num_pos    = (int*)(wsf + (size_t)4 * NBP + B_ * G_);
  float* accum      = wsf + (size_t)4 * NBP + B_ * G_ + B_;

  k_init  <<<1, 128, 0, stream>>>(num_pos, accum);
  k_assign<<<B_, 256, 0, stream>>>(priors, gt_box, best_iou, best_g, best_prior);
  k_encode<<<NBP / 256, 256, 0, stream>>>(loc_p, priors, gt_box, gt_class,
                                          best_iou, best_g, best_prior,
                                          class_t, num_pos, accum);
  k_ce    <<<NBP / 8, 256, 0, stream>>>(class_p, class_t, ce, accum);
  k_mine  <<<B_, 256, 0, stream>>>(ce, class_t, num_pos, accum);
  k_final <<<1, 1, 0, stream>>>(num_pos, accum, out);
}